// SynchPCFG_67602785239516
// MI455X (gfx1250) — compile-verified
//
#include <hip/hip_runtime.h>
#include <math.h>
#include <stdint.h>

// ---------------------------------------------------------------------------
// SynchPCFG terms kernel for gfx1250 (MI455X), fp32 WMMA path.
// ---------------------------------------------------------------------------

typedef __attribute__((ext_vector_type(2))) float v2f;
typedef __attribute__((ext_vector_type(8))) float v8f;

#define NEGV (-100000.0f)

namespace {
constexpr int H  = 256;
constexpr int VV = 16384;
constexpr int TT = 16;
constexpr int BB = 4;
constexpr int NN = 48;
constexpr int MM = 48;
constexpr int RT = BB * NN * TT; // 3072 rows for the tgt MLP
}

#define WMMA_F32(a, b, c) \
  __builtin_amdgcn_wmma_f32_16x16x4_f32(false, (a), false, (b), (short)0, (c), false, false)

// C/D 16x16 f32 layout: VGPR v, lanes 0-15 -> (M=v, N=lane); lanes 16-31 -> (M=v+8).
__device__ __forceinline__ void store_tile(float* __restrict__ Y, int ldy,
                                           int row0, int col0, v8f c,
                                           const float* __restrict__ bias,
                                           int act,
                                           const float* __restrict__ resid) {
  const int lane = threadIdx.x & 31;
  const int half = lane >> 4;
  const int n    = lane & 15;
  const float bv = bias ? bias[col0 + n] : 0.0f;
#pragma unroll
  for (int v = 0; v < 8; ++v) {
    const int r = row0 + v + 8 * half;
    float val = c[v] + bv;
    if (act) val = fmaxf(val, 0.0f);
    if (resid) val += resid[(size_t)r * ldy + col0 + n];
    Y[(size_t)r * ldy + col0 + n] = val;
  }
}

// ----- generic (R x 256) @ (256 x 256)^T GEMM with fused bias/ReLU/residual.
// grid (R/16), block 128 (4 waves); wave w owns a 16x64 output strip
// (4 col tiles), reusing one A fragment across 4 WMMAs per K step.
__global__ void gemm256_kernel(const float* __restrict__ X,
                               const float* __restrict__ W,
                               const float* __restrict__ bias,
                               const float* __restrict__ resid,
                               float* __restrict__ Y, int R, int act) {
  const int wave = threadIdx.x >> 5;
  const int lane = threadIdx.x & 31;
  const int half = lane >> 4;
  const int m    = lane & 15;
  const int row0 = blockIdx.x * 16;
  const int colb = wave * 64;

  const float* xr = X + (size_t)(row0 + m) * H + 2 * half;
  const float* wr = W + (size_t)(colb + m) * H + 2 * half;

  v8f cc[4];
#pragma unroll
  for (int j = 0; j < 4; ++j) cc[j] = (v8f){};

#pragma unroll 4
  for (int k = 0; k < H; k += 4) {
    v2f a;
    a.x = xr[k];
    a.y = xr[k + 1];
#pragma unroll
    for (int j = 0; j < 4; ++j) {
      v2f b;
      b.x = wr[(size_t)j * 16 * H + k];
      b.y = wr[(size_t)j * 16 * H + k + 1];
      cc[j] = WMMA_F32(a, b, cc[j]);
    }
  }
#pragma unroll
  for (int j = 0; j < 4; ++j)
    store_tile(Y, H, row0, colb + j * 16, cc[j], bias, act, resid);
}

// ----- streaming logits + online logsumexp over V columns.
// grid R/16, block 256 (8 waves); each wave strides over groups of 4 col tiles.
// X tile staged to LDS via async global->LDS copy (gfx1250 path).
__global__ void lse_kernel(const float* __restrict__ Xh,
                           const float* __restrict__ W,
                           const float* __restrict__ bias,
                           float* __restrict__ lse, int R, int Vtot) {
  __shared__ float xs[16 * H];
  __shared__ float pm[8][16];
  __shared__ float ps[8][16];
  const int row0 = blockIdx.x * 16;

  // async-stage 16KB X tile: each of 256 threads copies 64B (4 x b128).
  {
    const float* gsrc = Xh + (size_t)row0 * H + threadIdx.x * 16;
    unsigned lds_off = (unsigned)(uintptr_t)(xs + threadIdx.x * 16);
    asm volatile(
        "global_load_async_to_lds_b128 %0, %1, off\n\t"
        "global_load_async_to_lds_b128 %0, %1, off offset:16\n\t"
        "global_load_async_to_lds_b128 %0, %1, off offset:32\n\t"
        "global_load_async_to_lds_b128 %0, %1, off offset:48\n\t"
        "s_wait_asynccnt 0x0"
        :: "v"(lds_off), "v"(gsrc)
        : "memory");
  }
  __syncthreads();

  const int wave = threadIdx.x >> 5;
  const int lane = threadIdx.x & 31;
  const int half = lane >> 4;
  const int n    = lane & 15;
  const int nwaves  = blockDim.x >> 5;
  const int ngroups = Vtot / 64;   // 4 col tiles per group

  float mx[8], s[8];
#pragma unroll
  for (int v = 0; v < 8; ++v) { mx[v] = -INFINITY; s[v] = 0.0f; }

  const float* xr = xs + (size_t)n * H + 2 * half;

  for (int g = wave; g < ngroups; g += nwaves) {
    const int col0 = g * 64;
    const float* wr = W + (size_t)(col0 + n) * H + 2 * half;

    v8f cc[4];
#pragma unroll
    for (int j = 0; j < 4; ++j) cc[j] = (v8f){};

#pragma unroll 2
    for (int k = 0; k < H; k += 4) {
      v2f a;
      a.x = xr[k];
      a.y = xr[k + 1];
#pragma unroll
      for (int j = 0; j < 4; ++j) {
        v2f b;
        b.x = wr[(size_t)j * 16 * H + k];
        b.y = wr[(size_t)j * 16 * H + k + 1];
        cc[j] = WMMA_F32(a, b, cc[j]);
      }
    }

    // prefetch this lane's W row for the next group (global_prefetch_b8)
    if (col0 + nwaves * 64 < Vtot)
      __builtin_prefetch(W + (size_t)(col0 + nwaves * 64 + n) * H, 0, 1);

#pragma unroll
    for (int j = 0; j < 4; ++j) {
      const float bv = bias[col0 + j * 16 + n];
#pragma unroll
      for (int v = 0; v < 8; ++v) {
        const float val = cc[j][v] + bv;
        // row-wise reduce across the 16 lanes of this half (xor 1..8 keeps half)
        float bm = val;
        for (int d = 1; d < 16; d <<= 1) bm = fmaxf(bm, __shfl_xor(bm, d, 32));
        float e = __expf(val - bm);
        for (int d = 1; d < 16; d <<= 1) e += __shfl_xor(e, d, 32);
        const float nm = fmaxf(mx[v], bm);
        s[v] = s[v] * __expf(mx[v] - nm) + e * __expf(bm - nm);
        mx[v] = nm;
      }
    }
  }
  if (n == 0) {
#pragma unroll
    for (int v = 0; v < 8; ++v) {
      pm[wave][8 * half + v] = mx[v];
      ps[wave][8 * half + v] = s[v];
    }
  }
  __syncthreads();
  if (threadIdx.x < 16) {
    const int r = threadIdx.x;
    float m = -INFINITY;
    for (int w = 0; w < 8; ++w) m = fmaxf(m, pm[w][r]);
    float ss = 0.0f;
    for (int w = 0; w < 8; ++w) ss += ps[w][r] * __expf(pm[w][r] - m);
    lse[row0 + r] = m + __logf(ss);
  }
}

// ----- build tgt-MLP input rows: X0[(b*48+n)*16+t] = src_enc[src_ids[b,n]] + t_emb[t]
__global__ void build_x0_kernel(const float* __restrict__ src_enc,
                                const float* __restrict__ t_emb,
                                const int* __restrict__ src_ids,
                                float* __restrict__ X0) {
  const int row = blockIdx.x;
  const int bn = row >> 4;
  const int t  = row & 15;
  const int id = src_ids[bn];
  const int h = threadIdx.x;
  X0[(size_t)row * H + h] = src_enc[(size_t)id * H + h] + t_emb[(size_t)t * H + h];
}

// ----- tgt gathered log-probs: out[(b,n,t),m] = <hid[row], wout[id]> + b[id] - lse[row]
__global__ void tgt_gather_kernel(const float* __restrict__ Xh,
                                  const float* __restrict__ W,
                                  const float* __restrict__ bias,
                                  const int* __restrict__ tgt_ids,
                                  const float* __restrict__ lse,
                                  float* __restrict__ out) {
  const int row = blockIdx.x;         // (b*48+n)*16+t
  const int b = row / (NN * TT);
  const int wave = threadIdx.x >> 5;
  const int lane = threadIdx.x & 31;
  const float* x = Xh + (size_t)row * H;
  for (int m = wave; m < MM; m += 8) {
    const int col = tgt_ids[b * MM + m];
    const float* w = W + (size_t)col * H;
    float acc = 0.0f;
    for (int k = lane; k < H; k += 32) acc += x[k] * w[k];
    for (int d = 1; d < 32; d <<= 1) acc += __shfl_xor(acc, d, 32);
    if (lane == 0) out[(size_t)row * MM + m] = acc + bias[col] - lse[row];
  }
}

// ----- small-MLP gathered log-probs: out[j][t] = <hid[t], wout[ids[j]]> + b - lse[t]
__global__ void small_gather_kernel(const float* __restrict__ Xh,
                                    const float* __restrict__ W,
                                    const float* __restrict__ bias,
                                    const int* __restrict__ ids,
                                    const float* __restrict__ lse,
                                    float* __restrict__ out) {
  const int j = blockIdx.x;           // 0..191
  const int wave = threadIdx.x >> 5;
  const int lane = threadIdx.x & 31;
  const int col = ids[j];
  const float* w = W + (size_t)col * H;
  for (int t = wave; t < TT; t += 8) {
    const float* x = Xh + (size_t)t * H;
    float acc = 0.0f;
    for (int k = lane; k < H; k += 32) acc += x[k] * w[k];
    for (int d = 1; d < 32; d <<= 1) acc += __shfl_xor(acc, d, 32);
    if (lane == 0) out[(size_t)j * TT + t] = acc + bias[col] - lse[t];
  }
}

// ----- et_lp: log_softmax(t_type_emb @ et_w.T + et_b) -> (16,4)
__global__ void et_kernel(const float* __restrict__ t_type_emb,
                          const float* __restrict__ et_w,
                          const float* __restrict__ et_b,
                          float* __restrict__ et_lp) {
  __shared__ float lg[TT][4];
  const int t = threadIdx.x >> 2;
  const int c = threadIdx.x & 3;
  float acc = et_b[c];
  for (int k = 0; k < H; ++k) acc += t_type_emb[t * H + k] * et_w[c * H + k];
  lg[t][c] = acc;
  __syncthreads();
  float m = fmaxf(fmaxf(lg[t][0], lg[t][1]), fmaxf(lg[t][2], lg[t][3]));
  float ssum = __expf(lg[t][0] - m) + __expf(lg[t][1] - m) +
               __expf(lg[t][2] - m) + __expf(lg[t][3] - m);
  et_lp[t * 4 + c] = lg[t][c] - (m + __logf(ssum));
}

// ----- final assembly into terms (B,N,M,2,2,T)
__global__ void assemble_kernel(const float* __restrict__ src_g,  // (192,16)
                                const float* __restrict__ we_g,   // (192,16)
                                const float* __restrict__ ew_g,   // (192,16)
                                const float* __restrict__ tgt_terms, // (B,N,T,M)
                                const float* __restrict__ et_lp,  // (16,4)
                                const int* __restrict__ src_ids,
                                const int* __restrict__ tgt_ids,
                                float* __restrict__ out) {
  const int idx = blockIdx.x;               // ((b*48+n)*48+m)
  const int b = idx / (NN * MM);
  const int rem = idx % (NN * MM);
  const int n = rem / MM;
  const int m = rem % MM;
  const int q = threadIdx.x >> 4;           // 0:t00 1:t01 2:t10 3:t11
  const int t = threadIdx.x & 15;
  float val;
  if (q == 0) {
    val = NEGV;
  } else if (q == 1) {
    val = ew_g[(size_t)(b * MM + m) * TT + t] + et_lp[t * 4 + 3];
  } else if (q == 2) {
    val = we_g[(size_t)(b * NN + n) * TT + t] + et_lp[t * 4 + 2];
  } else {
    const float st = src_g[(size_t)(b * NN + n) * TT + t];
    const float ww_lp =
        st + tgt_terms[((size_t)(b * NN + n) * TT + t) * MM + m] + et_lp[t * 4 + 1];
    const float cm = (src_ids[b * NN + n] != tgt_ids[b * MM + m]) ? NEGV : 0.0f;
    const float copy_lp = cm + et_lp[t * 4 + 0] + st;
    const float a = fmaxf(copy_lp, ww_lp);
    const float mn = fminf(copy_lp, ww_lp);
    val = a + log1pf(__expf(mn - a));
  }
  out[((size_t)idx * 4 + q) * TT + t] = val;
}

// ---------------------------------------------------------------------------
// Host orchestration
// ---------------------------------------------------------------------------

namespace {
// workspace float offsets
constexpr size_t OFF_X0   = 0;
constexpr size_t OFF_A    = OFF_X0 + (size_t)RT * H;
constexpr size_t OFF_B    = OFF_A + (size_t)RT * H;
constexpr size_t OFF_C    = OFF_B + (size_t)RT * H;
constexpr size_t OFF_SA   = OFF_C + (size_t)RT * H;
constexpr size_t OFF_SB   = OFF_SA + (size_t)TT * H;
constexpr size_t OFF_SC   = OFF_SB + (size_t)TT * H;
constexpr size_t OFF_LSET = OFF_SC + (size_t)TT * H;      // 3072
constexpr size_t OFF_LSES = OFF_LSET + RT;                // 16
constexpr size_t OFF_TGT  = OFF_LSES + TT;                // 3072*48
constexpr size_t OFF_SRCG = OFF_TGT + (size_t)RT * MM;    // 192*16
constexpr size_t OFF_WEG  = OFF_SRCG + (size_t)BB * NN * TT;
constexpr size_t OFF_EWG  = OFF_WEG + (size_t)BB * NN * TT;
constexpr size_t OFF_ETLP = OFF_EWG + (size_t)BB * MM * TT;
}

extern "C" void kernel_launch(void* const* d_in, const int* in_sizes, int n_in,
                              void* d_out, int out_size, void* d_ws,
                              size_t ws_size, hipStream_t stream) {
  (void)in_sizes; (void)n_in; (void)out_size; (void)ws_size;
  auto F = [&](int i) { return (const float*)d_in[i]; };
  const float* t_emb      = F(0);
  const float* t_type_emb = F(1);
  const float* src_enc    = F(2);
  const float* et_w       = F(3);
  const float* et_b       = F(4);
  const int* src_ids = (const int*)d_in[29];
  const int* tgt_ids = (const int*)d_in[30];
  float* out = (float*)d_out;

  float* ws = (float*)d_ws;
  float* X0   = ws + OFF_X0;
  float* Ab   = ws + OFF_A;
  float* Bbuf = ws + OFF_B;
  float* Cb   = ws + OFF_C;
  float* SA   = ws + OFF_SA;
  float* SB   = ws + OFF_SB;
  float* SC   = ws + OFF_SC;
  float* lseT = ws + OFF_LSET;
  float* lseS = ws + OFF_LSES;
  float* tgtT = ws + OFF_TGT;
  float* srcG = ws + OFF_SRCG;
  float* weG  = ws + OFF_WEG;
  float* ewG  = ws + OFF_EWG;
  float* etLP = ws + OFF_ETLP;

  auto gemm = [&](const float* X, const float* W, const float* bias,
                  const float* resid, float* Y, int R, int act) {
    gemm256_kernel<<<dim3(R / 16), 128, 0, stream>>>(X, W, bias, resid, Y, R, act);
  };
  // _res block: in -> t1 (uses t2 as scratch; clobbers in)
  auto res_block = [&](float* in, float* t1, float* t2, const float* rw,
                       const float* rb, int R) -> float* {
    gemm(in, rw + 4 * H * H, rb + 4 * H, nullptr, t1, R, 0);  // h1
    gemm(t1, rw + 0 * H * H, rb + 0 * H, nullptr, t2, R, 1);  // a = relu
    gemm(t2, rw + 2 * H * H, rb + 2 * H, t1, in, R, 1);       // h2 = h1 + relu(.)
    gemm(in, rw + 1 * H * H, rb + 1 * H, nullptr, t2, R, 1);  // c = relu
    gemm(t2, rw + 3 * H * H, rb + 3 * H, in, t1, R, 1);       // h3 = h2 + relu(.)
    return t1;
  };
  // full hidden MLP + streaming logsumexp; returns hidden pointer
  auto run_mlp = [&](const float* X0p, int R, int base, float* A_, float* B_,
                     float* C_, float* lse_out) -> float* {
    const float* win  = F(base + 0);
    const float* binp = F(base + 1);
    const float* rw   = F(base + 2);
    const float* rb   = F(base + 3);
    const float* wout = F(base + 4);
    const float* bout = F(base + 5);
    gemm(X0p, win, binp, nullptr, A_, R, 0);
    float* h1 = res_block(A_, B_, C_, rw, rb, R);                       // -> B_
    float* h2 = res_block(h1, A_, C_, rw + 5 * H * H, rb + 5 * H, R);   // -> A_
    lse_kernel<<<dim3(R / 16), 256, 0, stream>>>(h2, wout, bout, lse_out, R, VV);
    return h2;
  };

  // 0) et_lp
  et_kernel<<<1, 64, 0, stream>>>(t_type_emb, et_w, et_b, etLP);

  // 1) tgt branch (3072 rows)
  build_x0_kernel<<<RT, H, 0, stream>>>(src_enc, t_emb, src_ids, X0);
  float* hidT = run_mlp(X0, RT, 11, Ab, Bbuf, Cb, lseT);
  tgt_gather_kernel<<<RT, 256, 0, stream>>>(hidT, F(15), F(16), tgt_ids, lseT, tgtT);

  // 2) src / we / ew branches (16 rows each)
  float* hs = run_mlp(t_emb, TT, 5, SA, SB, SC, lseS);
  small_gather_kernel<<<BB * NN, 256, 0, stream>>>(hs, F(9), F(10), src_ids, lseS, srcG);

  float* hw = run_mlp(t_emb, TT, 17, SA, SB, SC, lseS);
  small_gather_kernel<<<BB * NN, 256, 0, stream>>>(hw, F(21), F(22), src_ids, lseS, weG);

  float* he = run_mlp(t_emb, TT, 23, SA, SB, SC, lseS);
  small_gather_kernel<<<BB * MM, 256, 0, stream>>>(he, F(27), F(28), tgt_ids, lseS, ewG);

  // 3) assemble terms (B,N,M,2,2,T)
  assemble_kernel<<<BB * NN * MM, 64, 0, stream>>>(srcG, weG, ewG, tgtT, etLP,
                                                   src_ids, tgt_ids, out);
}